// Dcls1d_23940147708110
// MI455X (gfx1250) — compile-verified
//
#include <hip/hip_runtime.h>

typedef __attribute__((ext_vector_type(2))) float v2f;
typedef __attribute__((ext_vector_type(8))) float v8f;
typedef __attribute__((ext_vector_type(4))) int   v4i;

#define N_BATCH 32
#define C_IN    256
#define O_OUT   256
#define KTAPS   7
#define LDIL    56            // dense dilated kernel length
#define LIN     4096
#define PAD     28
#define LOUT    4097          // 4096 + 2*28 - 56 + 1
#define KTOT    (C_IN * LDIL) // 14336 = GEMM K

#define KB      32            // K elements staged per chunk (8 WMMA k-steps)
#define O_TILE  128           // M per block
#define L_TILE  32            // N per block
#define A_PITCH (KB + 4)      // 36 floats: conflict-free & keeps 16B row alignment

#if defined(__has_builtin)
#  if __has_builtin(__builtin_amdgcn_global_load_async_to_lds_b128)
#    define USE_ASYNC_LDS 1
#  else
#    define USE_ASYNC_LDS 0
#  endif
#else
#  define USE_ASYNC_LDS 0
#endif

typedef __attribute__((address_space(1))) v4i v4i_gbl;   // printed as "__device__" by HIP clang
typedef __attribute__((address_space(3))) v4i v4i_lds;   // printed as "__shared__"

__device__ __forceinline__ void wait_async_zero() {
#if USE_ASYNC_LDS
#  if __has_builtin(__builtin_amdgcn_s_wait_asynccnt)
    __builtin_amdgcn_s_wait_asynccnt(0);
#  else
    asm volatile("s_wait_asynccnt 0x0" ::: "memory");
#  endif
#endif
}

// ---------------- pass 1: zero the dense kernel ----------------
__global__ void zero_kernel(float* __restrict__ p, int n) {
    int i = blockIdx.x * blockDim.x + threadIdx.x;
    if (i < n) p[i] = 0.0f;
}

// ---------------- pass 2: scatter taps with linear interpolation ----------------
__global__ void scatter_kernel(const float* __restrict__ w,
                               const float* __restrict__ P,
                               float* __restrict__ Kd) {
    int idx = blockIdx.x * blockDim.x + threadIdx.x;
    if (idx >= O_OUT * C_IN * KTAPS) return;
    int o   = idx / (C_IN * KTAPS);
    int rem = idx % (C_IN * KTAPS);
    int c   = rem / KTAPS;

    float wv  = w[idx];
    float pos = P[idx] + (float)(LDIL / 2);
    pos = fminf(fmaxf(pos, 0.0f), (float)(LDIL - 1));   // clamp border
    float lo   = floorf(pos);
    float frac = pos - lo;
    int loi = (int)lo;
    int hii = min(loi + 1, LDIL - 1);

    float* row = Kd + (size_t)o * KTOT + c * LDIL;
    atomicAdd(row + loi, wv * (1.0f - frac));
    atomicAdd(row + hii, wv * frac);
}

// ---------------- pass 3: implicit-GEMM conv via f32 WMMA ----------------
// A tile: Kd[o0:o0+128, kk:kk+32], row-major As[m][k] (pitch 36 floats).
__device__ __forceinline__ void stage_A(float (*As)[A_PITCH],
                                        const float* __restrict__ Kd,
                                        int tid, int o0, int kk) {
    const int m  = tid >> 1;           // 0..127
    const int s0 = (tid & 1) << 4;     // 0 or 16 (half-row of 16 floats)
    const float* src = Kd + (size_t)(o0 + m) * KTOT + kk + s0;
    float* dst = &As[m][s0];
    __builtin_prefetch(src + KB, 0, 0);        // global_prefetch_b8: next chunk
#if USE_ASYNC_LDS
    // 4 x 16B per thread, global -> LDS, ASYNCcnt-tracked (no VGPR round-trip).
#pragma unroll
    for (int i = 0; i < 4; ++i) {
        __builtin_amdgcn_global_load_async_to_lds_b128(
            (v4i_gbl*)(src + 4 * i), (v4i_lds*)(dst + 4 * i), 0, 0);
    }
#else
    float4 q0 = *reinterpret_cast<const float4*>(src + 0);
    float4 q1 = *reinterpret_cast<const float4*>(src + 4);
    float4 q2 = *reinterpret_cast<const float4*>(src + 8);
    float4 q3 = *reinterpret_cast<const float4*>(src + 12);
    *reinterpret_cast<float4*>(dst + 0)  = q0;
    *reinterpret_cast<float4*>(dst + 4)  = q1;
    *reinterpret_cast<float4*>(dst + 8)  = q2;
    *reinterpret_cast<float4*>(dst + 12) = q3;
#endif
}

// B tile: Bs[k][l] = x[n, c(k), l0 + l + t(k) - PAD], zero padded (needs predicate).
__device__ __forceinline__ void stage_B(float (*Bs)[L_TILE + 4],
                                        const float* __restrict__ xrow,
                                        int tid, int l0, int kk) {
    const int kb   = tid >> 3;            // 0..31
    const int col0 = (tid & 7) << 2;      // 0,4,...,28
    const int k    = kk + kb;
    const int c    = k / LDIL;
    const int toff = k - c * LDIL;
    const float* xr = xrow + (size_t)c * LIN;
    const int base = l0 + col0 + toff - PAD;
#pragma unroll
    for (int j = 0; j < 4; ++j) {
        int lg = base + j;
        float v = 0.0f;
        if (lg >= 0 && lg < LIN) v = xr[lg];
        Bs[kb][col0 + j] = v;
    }
}

__global__ __launch_bounds__(256)
void conv_wmma_kernel(const float* __restrict__ x,
                      const float* __restrict__ Kd,
                      const float* __restrict__ bias,
                      float* __restrict__ out) {
    __shared__ float As[2][O_TILE][A_PITCH];   // [buf][m][k]  36,864 B
    __shared__ float Bs[2][KB][L_TILE + 4];    // [buf][k][l]   9,216 B

    const int tid  = threadIdx.x;
    const int lane = tid & 31;
    const int wave = tid >> 5;

    const int l0 = blockIdx.x * L_TILE;
    const int o0 = blockIdx.y * O_TILE;
    const int n  = blockIdx.z;

    const int msub = (wave >> 1) * 32;   // wave's M offset in tile (0/32/64/96)
    const int lsub = (wave & 1) * 16;    // wave's N offset in tile (0/16)
    const int ml   = lane & 15;          // M/N index within 16
    const int kp   = (lane >> 4) << 1;   // K pair: lanes 0-15 -> K0,1; 16-31 -> K2,3

    const float* xrow = x + (size_t)n * C_IN * LIN;

    v8f acc0 = {};   // rows msub..msub+15
    v8f acc1 = {};   // rows msub+16..msub+31

    stage_A(As[0], Kd, tid, o0, 0);
    stage_B(Bs[0], xrow, tid, l0, 0);

#pragma unroll 1
    for (int kk = 0; kk < KTOT; kk += KB) {
        const int buf = (kk / KB) & 1;
        wait_async_zero();    // this wave's async fills of `buf` are in LDS
        __syncthreads();      // all waves' staged data visible; prev compute done
        if (kk + KB < KTOT) {
            stage_A(As[buf ^ 1], Kd, tid, o0, kk + KB);
            stage_B(Bs[buf ^ 1], xrow, tid, l0, kk + KB);
        }
#pragma unroll
        for (int ks = 0; ks < KB / 4; ++ks) {
            const int kb = ks * 4 + kp;
            v2f a0 = *reinterpret_cast<const v2f*>(&As[buf][msub + ml][kb]);
            v2f a1 = *reinterpret_cast<const v2f*>(&As[buf][msub + 16 + ml][kb]);
            v2f b;
            b.x = Bs[buf][kb    ][lsub + ml];
            b.y = Bs[buf][kb + 1][lsub + ml];
            acc0 = __builtin_amdgcn_wmma_f32_16x16x4_f32(
                false, a0, false, b, (short)0, acc0, false, false);
            acc1 = __builtin_amdgcn_wmma_f32_16x16x4_f32(
                false, a1, false, b, (short)0, acc1, false, false);
        }
    }

    // Epilogue: C/D layout -> lane L holds N = L%16, M = r + (L/16)*8 for VGPR r.
    const int lg = l0 + lsub + ml;
    if (lg < LOUT) {
        const int mrow = (lane >> 4) << 3;
        float* obase = out + (size_t)n * O_OUT * LOUT + lg;
#pragma unroll
        for (int r = 0; r < 8; ++r) {
            int o  = o0 + msub + mrow + r;
            obase[(size_t)o * LOUT] = acc0[r] + bias[o];
            int o2 = o + 16;
            obase[(size_t)o2 * LOUT] = acc1[r] + bias[o2];
        }
    }
}

extern "C" void kernel_launch(void* const* d_in, const int* in_sizes, int n_in,
                              void* d_out, int out_size, void* d_ws, size_t ws_size,
                              hipStream_t stream) {
    const float* x    = (const float*)d_in[0];
    const float* w    = (const float*)d_in[1];
    const float* P    = (const float*)d_in[2];
    const float* bias = (const float*)d_in[3];
    float* out = (float*)d_out;
    float* Kd  = (float*)d_ws;    // dense kernel [O_OUT][KTOT] f32 (~14.7 MB)

    const int kelems = O_OUT * KTOT;
    zero_kernel<<<(kelems + 255) / 256, 256, 0, stream>>>(Kd, kelems);

    const int taps = O_OUT * C_IN * KTAPS;
    scatter_kernel<<<(taps + 255) / 256, 256, 0, stream>>>(w, P, Kd);

    dim3 grid((LOUT + L_TILE - 1) / L_TILE, O_OUT / O_TILE, N_BATCH);
    conv_wmma_kernel<<<grid, 256, 0, stream>>>(x, Kd, bias, out);
}